// NeuralODEModel_25640954757503
// MI455X (gfx1250) — compile-verified
//
#include <hip/hip_runtime.h>

typedef float v2f __attribute__((ext_vector_type(2)));
typedef float v8f __attribute__((ext_vector_type(8)));

#define B_    8
#define N_    128
#define D1_   128
#define H_    256
#define TS_   10
#define DT_   (1.0f/1200.0f)
#define STEPS_ 120

#if __has_builtin(__builtin_amdgcn_tanhf)
#define TANH_F32(x) __builtin_amdgcn_tanhf(x)   // hardware V_TANH_F32 (CDNA5 trans op)
#else
#define TANH_F32(x) tanhf(x)                    // ocml fallback
#endif

// One workgroup per 16-row tile of the [1024 x 128] state matrix.
// 8 waves: wave w owns layer-1 output cols [32w,32w+32) and layer-2 cols [16w,16w+16).
// Weights live in VGPRs (256 per wave), state y and hidden h live in LDS.
__global__ __launch_bounds__(256, 1)
void node_euler_kernel(const float* __restrict__ z0,
                       const float* __restrict__ dis,
                       const float* __restrict__ W1,
                       const float* __restrict__ b1,
                       const float* __restrict__ W2,
                       const float* __restrict__ b2,
                       float* __restrict__ out)
{
    __shared__ float yt[16 * D1_];   // 8 KB  : current state tile, row-major
    __shared__ float ht[16 * H_];    // 16 KB : tanh hidden tile, row-major
    __shared__ float dts[16];        // original disappear_time per row (mask)

    const int tid   = threadIdx.x;
    const int lane  = tid & 31;
    const int w     = tid >> 5;      // wave id 0..7
    const int blk   = blockIdx.x;    // 0..63  (M tile)
    const int mhalf = lane >> 4;     // 0 or 1 (lane half -> K/M split per ISA layout)
    const int lan16 = lane & 15;

    // ---- init y0 tile: concat(z0, disappear_time) ----
    {
        int idx = tid * 8;           // 2048 elements / 256 threads
        int m   = idx >> 7;          // row in tile
        int d0  = idx & 127;
        int r   = blk * 16 + m;      // global row in [0,1024)
        #pragma unroll
        for (int j = 0; j < 8; ++j) {
            int d = d0 + j;
            yt[m * D1_ + d] = (d < D1_ - 1) ? z0[r * (D1_ - 1) + d] : dis[r];
        }
    }
    if (tid < 16) dts[tid] = dis[blk * 16 + tid];
    __syncthreads();

    // ---- load weight B-fragments into registers (stationary for all 1080 steps) ----
    // B-matrix 4x16 layout: VGPR0 = rows K=0 (lanes0-15) / K=2 (lanes16-31), VGPR1 = K=1 / K=3,
    // column N = lane%16 within the 16-wide N tile.
    v2f w1a[32], w1b[32], w2f[64];
    const int na = 16 * (2 * w)     + lan16;  // layer-1 N tile A column
    const int nb = 16 * (2 * w + 1) + lan16;  // layer-1 N tile B column
    const int n2 = 16 * w           + lan16;  // layer-2 N column
    #pragma unroll
    for (int kt = 0; kt < 32; ++kt) {
        int k0 = 4 * kt + 2 * mhalf;
        w1a[kt] = v2f{ W1[k0 * H_ + na], W1[(k0 + 1) * H_ + na] };
        w1b[kt] = v2f{ W1[k0 * H_ + nb], W1[(k0 + 1) * H_ + nb] };
    }
    #pragma unroll
    for (int kt = 0; kt < 64; ++kt) {
        int k0 = 4 * kt + 2 * mhalf;
        w2f[kt] = v2f{ W2[k0 * D1_ + n2], W2[(k0 + 1) * D1_ + n2] };
    }
    const float bias1a = b1[na];
    const float bias1b = b1[nb];
    const float bias2  = b2[n2];

    const int bIdx = blk >> 3;             // batch index
    const int n0   = (blk & 7) * 16;       // first N row of this tile within batch

    // ---- masked snapshot writer: out[b, ti, n, d] ----
    auto write_out = [&](int ti) {
        float tcur = (float)ti / 10.0f;
        int idx = tid * 8;
        int m   = idx >> 7;
        int d0  = idx & 127;
        float msk = (tcur < dts[m]) ? 1.0f : 0.0f;
        size_t base = (((size_t)bIdx * TS_ + ti) * N_ + (n0 + m)) * (size_t)D1_ + d0;
        #pragma unroll
        for (int j = 0; j < 8; ++j)
            out[base + j] = yt[m * D1_ + d0 + j] * msk;
    };

    write_out(0);

    const int mrow = 8 * mhalf;            // C/D layout: M = vgpr + 8*(lane/16)

    for (int chunk = 0; chunk < TS_ - 1; ++chunk) {
        for (int s = 0; s < STEPS_; ++s) {
            // ---------- layer 1: h = tanh(y @ W1 + b1) ----------
            // two independent accumulator chains (two N tiles) interleaved for ILP
            v8f acc0, acc1;
            #pragma unroll
            for (int v = 0; v < 8; ++v) { acc0[v] = bias1a; acc1[v] = bias1b; }
            #pragma unroll
            for (int kt = 0; kt < 32; ++kt) {
                // A-matrix 16x4 layout: lane%16 = M row, VGPR0/1 = K = 4kt+2*(lane/16) +0/+1
                v2f a = *(const v2f*)&yt[lan16 * D1_ + 4 * kt + 2 * mhalf];
                acc0 = __builtin_amdgcn_wmma_f32_16x16x4_f32(
                           false, a, false, w1a[kt], (short)0, acc0, false, false);
                acc1 = __builtin_amdgcn_wmma_f32_16x16x4_f32(
                           false, a, false, w1b[kt], (short)0, acc1, false, false);
            }
            #pragma unroll
            for (int v = 0; v < 8; ++v) {
                ht[(mrow + v) * H_ + na] = TANH_F32(acc0[v]);
                ht[(mrow + v) * H_ + nb] = TANH_F32(acc1[v]);
            }
            __syncthreads();

            // ---------- layer 2: y += DT * (h @ W2 + b2) ----------
            // split K into two independent accumulator chains to break the 64-deep
            // D->C dependency chain, then combine.
            v8f acc2a, acc2b;
            #pragma unroll
            for (int v = 0; v < 8; ++v) { acc2a[v] = bias2; acc2b[v] = 0.0f; }
            #pragma unroll
            for (int kt = 0; kt < 32; ++kt) {
                v2f aA = *(const v2f*)&ht[lan16 * H_ + 4 * kt        + 2 * mhalf];
                v2f aB = *(const v2f*)&ht[lan16 * H_ + 4 * (kt + 32) + 2 * mhalf];
                acc2a = __builtin_amdgcn_wmma_f32_16x16x4_f32(
                            false, aA, false, w2f[kt],      (short)0, acc2a, false, false);
                acc2b = __builtin_amdgcn_wmma_f32_16x16x4_f32(
                            false, aB, false, w2f[kt + 32], (short)0, acc2b, false, false);
            }
            #pragma unroll
            for (int v = 0; v < 8; ++v) {
                yt[(mrow + v) * D1_ + n2] += DT_ * (acc2a[v] + acc2b[v]);
            }
            __syncthreads();
        }
        write_out(chunk + 1);
    }
}

extern "C" void kernel_launch(void* const* d_in, const int* in_sizes, int n_in,
                              void* d_out, int out_size, void* d_ws, size_t ws_size,
                              hipStream_t stream) {
    const float* z0  = (const float*)d_in[0];   // [8,128,127]
    const float* dis = (const float*)d_in[1];   // [8,128,1]
    // d_in[2] = t, unused (mask uses ti/10 directly, matching reference semantics)
    const float* W1  = (const float*)d_in[3];   // [128,256]
    const float* b1  = (const float*)d_in[4];   // [256]
    const float* W2  = (const float*)d_in[5];   // [256,128]
    const float* b2  = (const float*)d_in[6];   // [128]
    float* out = (float*)d_out;                 // [8,10,128,128]

    node_euler_kernel<<<64, 256, 0, stream>>>(z0, dis, W1, b1, W2, b2, out);
}